// RelGraphConvLayer_2181843386580
// MI455X (gfx1250) — compile-verified
//
#include <hip/hip_runtime.h>

typedef float v2f __attribute__((ext_vector_type(2)));
typedef float v8f __attribute__((ext_vector_type(8)));

#define N_NODES 50000
#define N_REL   8
#define N_EDGES 400000
#define D_IN    256
#define D_OUT   256

// ---------------------------------------------------------------------------
// LDS helpers: 32-bit LDS byte offset of a __shared__ object (for async DMA)
// ---------------------------------------------------------------------------
__device__ __forceinline__ unsigned lds_off_u32(const void* p) {
    return (unsigned)(uintptr_t)(__attribute__((address_space(3))) const void*)p;
}

// Stage a 256x32 f32 weight tile (cols [n0, n0+32)) into LDS with the CDNA5
// async global->LDS path. Thread t copies W row k=t (32 floats = 8 x b128).
// Tracked by ASYNCcnt; drained with s_wait_asynccnt, then workgroup barrier.
__device__ __forceinline__ void stage_w_tile_async(const float* __restrict__ W,
                                                   int n0, float* Wtile) {
    const int t = threadIdx.x;                       // 0..255 -> k row
    const float* wsrc = W + (size_t)t * D_OUT + n0;  // 32 contiguous floats
    unsigned ldsoff = lds_off_u32(&Wtile[t * 32]);
    #pragma unroll
    for (int j = 0; j < 8; ++j) {
        asm volatile("global_load_async_to_lds_b128 %0, %1, off"
                     :: "v"(ldsoff + j * 16), "v"(wsrc + j * 4)
                     : "memory");
    }
    asm volatile("s_wait_asynccnt 0x0" ::: "memory");
    __syncthreads();
}

// ---------------------------------------------------------------------------
// Zero helpers
// ---------------------------------------------------------------------------
__global__ __launch_bounds__(256) void zero_f4_kernel(float4* __restrict__ p, int n4) {
    int i = blockIdx.x * 256 + threadIdx.x;
    if (i < n4) p[i] = make_float4(0.f, 0.f, 0.f, 0.f);
}

__global__ __launch_bounds__(256) void zero_i32_kernel(int* __restrict__ p, int n) {
    int i = blockIdx.x * 256 + threadIdx.x;
    if (i < n) p[i] = 0;
}

// ---------------------------------------------------------------------------
// Degree histogram over all relations: deg[r*N + dst[r][e]] += 1
// ---------------------------------------------------------------------------
__global__ __launch_bounds__(256) void degree_kernel(const int* __restrict__ dst,
                                                     int* __restrict__ deg) {
    int i = blockIdx.x * 256 + threadIdx.x;          // over R*E
    if (i >= N_REL * N_EDGES) return;
    int r = i / N_EDGES;
    atomicAdd(&deg[r * N_NODES + dst[i]], 1);
}

// ---------------------------------------------------------------------------
// H = X * W   (M x 256) * (256 x 256), f32 WMMA 16x16x4.
// Block = 256 threads = 8 waves; each wave owns a 16x16 output tile.
// Block tile: 64 rows x 32 cols -> grid ((M+63)/64, 256/32).
// W tile staged in LDS via async DMA; A fragments streamed from global (x is
// L2-resident: 51 MB << 192 MB L2).
// A frag (16x4 f32, 2 VGPR): lane l: m = m0+(l&15), k-pair = (l>>4)*2.
// B frag (4x16 f32, 2 VGPR): lane l: n = n0+(l&15), k-pair = (l>>4)*2.
// C/D (16x16 f32, 8 VGPR):   row = m0 + v + 8*(l>>4), col = n0+(l&15).
// ---------------------------------------------------------------------------
__global__ __launch_bounds__(256) void xw_gemm_kernel(const float* __restrict__ X,
                                                      const float* __restrict__ W,
                                                      float* __restrict__ H) {
    __shared__ float Wtile[D_IN * 32];               // 32 KB, row stride 32

    const int lane  = threadIdx.x & 31;
    const int wave  = threadIdx.x >> 5;
    const int m0    = blockIdx.x * 64 + (wave >> 1) * 16;
    const int n0    = blockIdx.y * 32 + (wave & 1) * 16;

    stage_w_tile_async(W, blockIdx.y * 32, Wtile);   // all waves participate

    if (m0 >= N_NODES) return;                       // whole 16-row tile OOB (50000 % 16 == 0)

    const int half  = lane >> 4;                     // 0: K={0,1}, 1: K={2,3}
    const int idx16 = lane & 15;
    const int kb    = half * 2;
    const int ncol  = (wave & 1) * 16 + idx16;       // column within the 32-wide LDS tile
    const float* xrow = X + (size_t)(m0 + idx16) * D_IN;

    v8f c = {};
    for (int k = 0; k < D_IN; k += 4) {
        v2f a, b;
        a.x = xrow[k + kb];
        a.y = xrow[k + kb + 1];
        b.x = Wtile[(k + kb)     * 32 + ncol];       // ds_load, conflict-free
        b.y = Wtile[(k + kb + 1) * 32 + ncol];
        c = __builtin_amdgcn_wmma_f32_16x16x4_f32(false, a, false, b,
                                                  (short)0, c, false, false);
    }

    float* hp = H + (size_t)(m0 + half * 8) * D_OUT + n0 + idx16;
    #pragma unroll
    for (int v = 0; v < 8; ++v) hp[(size_t)v * D_OUT] = c[v];
}

// ---------------------------------------------------------------------------
// Edge scatter: out[dst] += H[src] / max(deg[dst],1)
// One wave per edge; lane covers 8 of the 256 columns (coalesced).
// ---------------------------------------------------------------------------
__global__ __launch_bounds__(256) void scatter_kernel(const float* __restrict__ H,
                                                      const int* __restrict__ src,
                                                      const int* __restrict__ dst,
                                                      const int* __restrict__ deg,
                                                      float* __restrict__ out) {
    int e = blockIdx.x * 8 + (threadIdx.x >> 5);
    if (e >= N_EDGES) return;
    const int lane = threadIdx.x & 31;
    const int s = src[e];
    const int d = dst[e];
    int dg = deg[d];
    const float scale = 1.0f / (float)(dg < 1 ? 1 : dg);
    const float* hrow = H + (size_t)s * D_OUT;
    float* orow       = out + (size_t)d * D_OUT;
    #pragma unroll
    for (int j = 0; j < 8; ++j) {
        int col = lane + j * 32;
        unsafeAtomicAdd(&orow[col], hrow[col] * scale);   // global_atomic_add_f32 in L2
    }
}

// ---------------------------------------------------------------------------
// Final: out = relu(out + X * W_loop + bias). Same WMMA tiling as xw_gemm,
// but C is seeded from the accumulated `out` and the epilogue fuses bias+relu.
// ---------------------------------------------------------------------------
__global__ __launch_bounds__(256) void selfloop_gemm_kernel(const float* __restrict__ X,
                                                            const float* __restrict__ W,
                                                            const float* __restrict__ bias,
                                                            float* __restrict__ out) {
    __shared__ float Wtile[D_IN * 32];

    const int lane  = threadIdx.x & 31;
    const int wave  = threadIdx.x >> 5;
    const int m0    = blockIdx.x * 64 + (wave >> 1) * 16;
    const int n0    = blockIdx.y * 32 + (wave & 1) * 16;

    stage_w_tile_async(W, blockIdx.y * 32, Wtile);

    if (m0 >= N_NODES) return;

    const int half  = lane >> 4;
    const int idx16 = lane & 15;
    const int kb    = half * 2;
    const int ncol  = (wave & 1) * 16 + idx16;
    const float* xrow = X + (size_t)(m0 + idx16) * D_IN;

    float* op = out + (size_t)(m0 + half * 8) * D_OUT + n0 + idx16;

    v8f c;
    #pragma unroll
    for (int v = 0; v < 8; ++v) c[v] = op[(size_t)v * D_OUT];   // seed C with aggregated messages

    for (int k = 0; k < D_IN; k += 4) {
        v2f a, b;
        a.x = xrow[k + kb];
        a.y = xrow[k + kb + 1];
        b.x = Wtile[(k + kb)     * 32 + ncol];
        b.y = Wtile[(k + kb + 1) * 32 + ncol];
        c = __builtin_amdgcn_wmma_f32_16x16x4_f32(false, a, false, b,
                                                  (short)0, c, false, false);
    }

    const float bn = bias[n0 + idx16];
    #pragma unroll
    for (int v = 0; v < 8; ++v) op[(size_t)v * D_OUT] = fmaxf(c[v] + bn, 0.0f);
}

// ---------------------------------------------------------------------------
// Host orchestration
// ---------------------------------------------------------------------------
extern "C" void kernel_launch(void* const* d_in, const int* in_sizes, int n_in,
                              void* d_out, int out_size, void* d_ws, size_t ws_size,
                              hipStream_t stream) {
    const float* x      = (const float*)d_in[0];   // [N, 256]
    const float* weight = (const float*)d_in[1];   // [8, 256, 256]
    const float* h_bias = (const float*)d_in[2];   // [256]
    const float* loop_w = (const float*)d_in[3];   // [256, 256]
    const int*   src    = (const int*)d_in[4];     // [8, E]
    const int*   dst    = (const int*)d_in[5];     // [8, E]
    float* out = (float*)d_out;                    // [N, 256]

    // Workspace layout: H [N*256 f32] | deg [8*N i32]   (~52.8 MB)
    float* H  = (float*)d_ws;
    int* deg  = (int*)((char*)d_ws + (size_t)N_NODES * D_OUT * sizeof(float));

    // 1) zero output accumulator + degree buffer
    {
        int n4 = (N_NODES * D_OUT) / 4;
        zero_f4_kernel<<<(n4 + 255) / 256, 256, 0, stream>>>((float4*)out, n4);
        int nd = N_REL * N_NODES;
        zero_i32_kernel<<<(nd + 255) / 256, 256, 0, stream>>>(deg, nd);
    }

    // 2) degree histogram for all relations
    {
        int tot = N_REL * N_EDGES;
        degree_kernel<<<(tot + 255) / 256, 256, 0, stream>>>(dst, deg);
    }

    // 3) per relation: transform-first GEMM then normalized edge scatter
    dim3 ggrid((N_NODES + 63) / 64, D_OUT / 32);
    for (int r = 0; r < N_REL; ++r) {
        xw_gemm_kernel<<<ggrid, 256, 0, stream>>>(x, weight + (size_t)r * D_IN * D_OUT, H);
        scatter_kernel<<<(N_EDGES + 7) / 8, 256, 0, stream>>>(
            H, src + (size_t)r * N_EDGES, dst + (size_t)r * N_EDGES,
            deg + (size_t)r * N_NODES, out);
    }

    // 4) self-loop GEMM + bias + relu (reads accumulator as C)
    selfloop_gemm_kernel<<<ggrid, 256, 0, stream>>>(x, loop_w, h_bias, out);

    (void)in_sizes; (void)n_in; (void)out_size; (void)ws_size;
}